// GATConv_layer_72688026518087
// MI455X (gfx1250) — compile-verified
//
#include <hip/hip_runtime.h>
#include <math.h>

// Problem constants (match reference)
#define NN 50000
#define EE 800000
#define HH 4
#define CC 64
#define HC 256        // H*C
#define DE 16
#define GG 64
#define NEG_SLOPE 0.2f
#define GN_EPS 1e-5f

typedef __attribute__((ext_vector_type(2))) float v2f;
typedef __attribute__((ext_vector_type(8))) float v8f;

// ---------------------------------------------------------------------------
// Init all accumulators (runs every launch -> deterministic under graph replay)
// ---------------------------------------------------------------------------
__global__ void __launch_bounds__(256) k_init(float* __restrict__ amax,
                                              float* __restrict__ denom,
                                              float* __restrict__ accum,
                                              float* __restrict__ gsum,
                                              float* __restrict__ gvar,
                                              float* __restrict__ cnt) {
  int idx = blockIdx.x * blockDim.x + threadIdx.x;
  if (idx < NN * HH) { amax[idx] = -INFINITY; denom[idx] = 0.f; }
  if (idx < NN * CC) accum[idx] = 0.f;
  if (idx < GG * CC) { gsum[idx] = 0.f; gvar[idx] = 0.f; }
  if (idx < GG) cnt[idx] = 0.f;
}

// ---------------------------------------------------------------------------
// Fold W_edge with att_edge:  V[d,h] = sum_c W_edge[d, h*64+c] * att_edge[h,c]
// ---------------------------------------------------------------------------
__global__ void __launch_bounds__(64) k_edge_vec(const float* __restrict__ W_edge,
                                                 const float* __restrict__ att_edge,
                                                 float* __restrict__ V) {
  int t = threadIdx.x;           // 64 threads: (d, h)
  int d = t >> 2, h = t & 3;
  float s = 0.f;
  for (int c = 0; c < CC; ++c)
    s += W_edge[d * HC + h * CC + c] * att_edge[h * CC + c];
  V[d * HH + h] = s;
}

// ---------------------------------------------------------------------------
// x = node(50000x64) @ W(64x256) via V_WMMA_F32_16X16X4_F32
// One wave per 16x16 output tile; K loop of 16 WMMAs.
// fp32 A 16x4 layout : lanes 0-15 = rows M, VGPR0 holds K+0 (lo half)/K+2 (hi),
//                      VGPR1 holds K+1 / K+3.
// fp32 B 4x16 layout : VGPR0 row K+0 (lanes 0-15) / K+2 (lanes 16-31), VGPR1 K+1/K+3.
// fp32 D 16x16      : VGPR r = row m0+r (lanes 0-15) / m0+r+8 (lanes 16-31).
// ---------------------------------------------------------------------------
__global__ void __launch_bounds__(256) k_proj_wmma(const float* __restrict__ node,
                                                   const float* __restrict__ W,
                                                   float* __restrict__ x) {
  const int wave  = (blockIdx.x * blockDim.x + threadIdx.x) >> 5;
  const int lane  = threadIdx.x & 31;
  const int tilesN = HC / 16;                  // 16
  const int m0 = (wave / tilesN) * 16;
  const int n0 = (wave % tilesN) * 16;
  if (m0 >= NN) return;                        // never true (50000 % 16 == 0), uniform

  const int lhalf = lane & 15;
  const int khi   = (lane >> 4) ? 2 : 0;

  v8f acc = {};
  #pragma unroll
  for (int k = 0; k < CC; k += 4) {
    const float* ap = node + (size_t)(m0 + lhalf) * CC + (k + khi);
    v2f a; a.x = ap[0]; a.y = ap[1];
    const float* bp = W + (size_t)(k + khi) * HC + n0 + lhalf;
    v2f b; b.x = bp[0]; b.y = bp[HC];
    acc = __builtin_amdgcn_wmma_f32_16x16x4_f32(false, a, false, b,
                                                (short)0, acc, false, false);
  }
  const int row_off = (lane >> 4) ? 8 : 0;
  #pragma unroll
  for (int r = 0; r < 8; ++r)
    x[(size_t)(m0 + r + row_off) * HC + n0 + lhalf] = acc[r];
}

// ---------------------------------------------------------------------------
// Per-node attention logits: a_src[n,h] = x[n,h,:] . att_src[h,:]  (same for dst)
// One 64-thread block per node, shared-memory tree reduction.
// ---------------------------------------------------------------------------
__global__ void __launch_bounds__(64) k_node_logits(const float* __restrict__ x,
                                                    const float* __restrict__ att_src,
                                                    const float* __restrict__ att_dst,
                                                    float* __restrict__ a_src,
                                                    float* __restrict__ a_dst) {
  __shared__ float red[2][HH][CC];
  const int n = blockIdx.x;
  const int c = threadIdx.x;
  const float* xr = x + (size_t)n * HC;
  #pragma unroll
  for (int h = 0; h < HH; ++h) {
    float v = xr[h * CC + c];
    red[0][h][c] = v * att_src[h * CC + c];
    red[1][h][c] = v * att_dst[h * CC + c];
  }
  __syncthreads();
  for (int s = 32; s > 0; s >>= 1) {
    if (c < s) {
      #pragma unroll
      for (int h = 0; h < HH; ++h) {
        red[0][h][c] += red[0][h][c + s];
        red[1][h][c] += red[1][h][c + s];
      }
    }
    __syncthreads();
  }
  if (c < HH) {
    a_src[n * HH + c] = red[0][c][0];
    a_dst[n * HH + c] = red[1][c][0];
  }
}

// float atomic-max via sign-split int/uint atomics (amax init = -inf)
__device__ __forceinline__ void atomicMaxF(float* addr, float val) {
  if (val >= 0.f) atomicMax((int*)addr, __float_as_int(val));
  else            atomicMin((unsigned int*)addr, (unsigned int)__float_as_int(val));
}

// ---------------------------------------------------------------------------
// alpha[e,h] = leaky_relu(a_src[src]+a_dst[dst]+edge_attr[e].V[:,h]); seg-max
// ---------------------------------------------------------------------------
__global__ void __launch_bounds__(256) k_alpha_max(const float* __restrict__ edge_attr,
                                                   const int* __restrict__ src_idx,
                                                   const int* __restrict__ dst_idx,
                                                   const float* __restrict__ a_src,
                                                   const float* __restrict__ a_dst,
                                                   const float* __restrict__ V,
                                                   float* __restrict__ alpha,
                                                   float* __restrict__ amax) {
  __shared__ float sV[DE * HH];
  if (threadIdx.x < DE * HH) sV[threadIdx.x] = V[threadIdx.x];
  __syncthreads();
  int e = blockIdx.x * blockDim.x + threadIdx.x;
  if (e >= EE) return;
  int s = src_idx[e], d = dst_idx[e];
  float acc[HH] = {0.f, 0.f, 0.f, 0.f};
  const float4* ep = (const float4*)(edge_attr + (size_t)e * DE);
  #pragma unroll
  for (int q = 0; q < 4; ++q) {
    float4 v4 = ep[q];
    float vv[4] = {v4.x, v4.y, v4.z, v4.w};
    #pragma unroll
    for (int j = 0; j < 4; ++j)
      #pragma unroll
      for (int h = 0; h < HH; ++h)
        acc[h] += vv[j] * sV[(q * 4 + j) * HH + h];
  }
  #pragma unroll
  for (int h = 0; h < HH; ++h) {
    float al = a_src[s * HH + h] + a_dst[d * HH + h] + acc[h];
    al = al > 0.f ? al : NEG_SLOPE * al;
    alpha[(size_t)e * HH + h] = al;
    atomicMaxF(&amax[d * HH + h], al);
  }
}

// ---------------------------------------------------------------------------
// ex = exp(alpha - amax[dst]) (in place); denom[dst,h] += ex
// ---------------------------------------------------------------------------
__global__ void __launch_bounds__(256) k_exp_sum(const int* __restrict__ dst_idx,
                                                 const float* __restrict__ amax,
                                                 float* __restrict__ alpha,
                                                 float* __restrict__ denom) {
  int idx = blockIdx.x * blockDim.x + threadIdx.x;   // over E*H
  if (idx >= EE * HH) return;
  int e = idx >> 2, h = idx & 3;
  int d = dst_idx[e];
  float m = amax[d * HH + h];
  if (!isfinite(m)) m = 0.f;                         // matches reference guard
  float ex = __expf(alpha[idx] - m);
  alpha[idx] = ex;
  atomicAdd(&denom[d * HH + h], ex);
}

// ---------------------------------------------------------------------------
// att = ex/(denom+1e-16); write att output; gather x[src] and scatter-add
// sum_h x[src,h,c]*att[e,h] into accum[dst,c].   4 edges per 256-thread block.
// ---------------------------------------------------------------------------
__global__ void __launch_bounds__(256) k_att_scatter(const int* __restrict__ src_idx,
                                                     const int* __restrict__ dst_idx,
                                                     const float* __restrict__ ex,
                                                     const float* __restrict__ denom,
                                                     const float* __restrict__ x,
                                                     float* __restrict__ att_out,
                                                     float* __restrict__ accum) {
  __shared__ float satt[4][HH];
  int le = threadIdx.x >> 6;                 // local edge 0..3
  int c  = threadIdx.x & 63;
  int e  = blockIdx.x * 4 + le;              // E % 4 == 0
  int s  = src_idx[e], d = dst_idx[e];
  if (c < HH) {
    float a = ex[(size_t)e * HH + c] / (denom[d * HH + c] + 1e-16f);
    satt[le][c] = a;
    att_out[(size_t)e * HH + c] = a;
  }
  __syncthreads();
  const float* xr = x + (size_t)s * HC;
  float acc = 0.f;
  #pragma unroll
  for (int h = 0; h < HH; ++h) acc += xr[h * CC + c] * satt[le][h];
  atomicAdd(&accum[(size_t)d * CC + c], acc);
}

// ---------------------------------------------------------------------------
// out = accum/H + bias (in place); group sums + counts
// ---------------------------------------------------------------------------
__global__ void __launch_bounds__(256) k_mean_bias(float* __restrict__ accum,
                                                   const float* __restrict__ bias,
                                                   const int* __restrict__ batch_ptr,
                                                   float* __restrict__ gsum,
                                                   float* __restrict__ cnt) {
  int idx = blockIdx.x * blockDim.x + threadIdx.x;
  if (idx >= NN * CC) return;
  int n = idx >> 6, c = idx & 63;
  float v = accum[idx] * (1.0f / HH) + bias[c];
  accum[idx] = v;
  int g = batch_ptr[n];
  atomicAdd(&gsum[g * CC + c], v);
  if (c == 0) atomicAdd(&cnt[g], 1.0f);
}

__global__ void __launch_bounds__(256) k_group_mean(float* __restrict__ gsum,
                                                    const float* __restrict__ cnt) {
  int idx = blockIdx.x * blockDim.x + threadIdx.x;   // G*C
  if (idx >= GG * CC) return;
  float cc = cnt[idx >> 6];
  gsum[idx] /= (cc > 1.f ? cc : 1.f);                // now holds mean
}

__global__ void __launch_bounds__(256) k_center_var(float* __restrict__ accum,
                                                    const float* __restrict__ gmean,
                                                    const float* __restrict__ gn_scale,
                                                    const int* __restrict__ batch_ptr,
                                                    float* __restrict__ gvar) {
  int idx = blockIdx.x * blockDim.x + threadIdx.x;
  if (idx >= NN * CC) return;
  int n = idx >> 6, c = idx & 63;
  int g = batch_ptr[n];
  float o = accum[idx] - gmean[g * CC + c] * gn_scale[c];
  accum[idx] = o;
  atomicAdd(&gvar[g * CC + c], o * o);
}

__global__ void __launch_bounds__(256) k_group_inv(float* __restrict__ gvar,
                                                   const float* __restrict__ cnt) {
  int idx = blockIdx.x * blockDim.x + threadIdx.x;   // G*C
  if (idx >= GG * CC) return;
  float cc = cnt[idx >> 6];
  cc = cc > 1.f ? cc : 1.f;
  gvar[idx] = rsqrtf(gvar[idx] / cc + GN_EPS);       // now holds 1/sqrt(var+eps)
}

__global__ void __launch_bounds__(256) k_norm_out(const float* __restrict__ accum,
                                                  const float* __restrict__ ginv,
                                                  const float* __restrict__ gn_w,
                                                  const float* __restrict__ gn_b,
                                                  const int* __restrict__ batch_ptr,
                                                  float* __restrict__ y) {
  int idx = blockIdx.x * blockDim.x + threadIdx.x;
  if (idx >= NN * CC) return;
  int n = idx >> 6, c = idx & 63;
  int g = batch_ptr[n];
  float v = gn_w[c] * (accum[idx] * ginv[g * CC + c]) + gn_b[c];
  y[idx] = v > 0.f ? v : 0.f;
}

// ---------------------------------------------------------------------------
extern "C" void kernel_launch(void* const* d_in, const int* in_sizes, int n_in,
                              void* d_out, int out_size, void* d_ws, size_t ws_size,
                              hipStream_t stream) {
  const float* node      = (const float*)d_in[0];
  const int*   edge_idx  = (const int*)d_in[1];      // [2,E]: src then dst
  const float* edge_attr = (const float*)d_in[2];
  const int*   batch_ptr = (const int*)d_in[3];
  const float* W         = (const float*)d_in[4];
  const float* W_edge    = (const float*)d_in[5];
  const float* att_src   = (const float*)d_in[6];
  const float* att_dst   = (const float*)d_in[7];
  const float* att_edge  = (const float*)d_in[8];
  const float* bias      = (const float*)d_in[9];
  const float* gn_w      = (const float*)d_in[10];
  const float* gn_b      = (const float*)d_in[11];
  const float* gn_scale  = (const float*)d_in[12];

  const int* src_idx = edge_idx;
  const int* dst_idx = edge_idx + EE;

  float* y_out   = (float*)d_out;                    // [N*C]
  float* att_out = y_out + (size_t)NN * CC;          // [E*H]

  // workspace layout (floats)
  float* ws    = (float*)d_ws;
  float* x     = ws;                                 // N*HC      = 12,800,000
  float* a_src = x     + (size_t)NN * HC;            // N*H       =    200,000
  float* a_dst = a_src + (size_t)NN * HH;            // N*H
  float* alpha = a_dst + (size_t)NN * HH;            // E*H       =  3,200,000
  float* amax  = alpha + (size_t)EE * HH;            // N*H
  float* denom = amax  + (size_t)NN * HH;            // N*H
  float* accum = denom + (size_t)NN * HH;            // N*C       =  3,200,000
  float* gsum  = accum + (size_t)NN * CC;            // G*C
  float* gvar  = gsum  + (size_t)GG * CC;            // G*C
  float* cnt   = gvar  + (size_t)GG * CC;            // G
  float* Vbuf  = cnt   + GG;                         // D_EDGE*H = 64

  const int blkNC = (NN * CC + 255) / 256;           // 12500
  const int blkEH = (EE * HH + 255) / 256;           // 12500
  const int blkE  = (EE + 255) / 256;                // 3125
  const int blkGC = (GG * CC + 255) / 256;           // 16

  k_init<<<blkNC, 256, 0, stream>>>(amax, denom, accum, gsum, gvar, cnt);
  k_edge_vec<<<1, 64, 0, stream>>>(W_edge, att_edge, Vbuf);

  // WMMA projection: 50000 tiles, 8 waves/block -> 6250 blocks
  k_proj_wmma<<<(NN / 16) * (HC / 16) / 8, 256, 0, stream>>>(node, W, x);

  k_node_logits<<<NN, 64, 0, stream>>>(x, att_src, att_dst, a_src, a_dst);
  k_alpha_max<<<blkE, 256, 0, stream>>>(edge_attr, src_idx, dst_idx,
                                        a_src, a_dst, Vbuf, alpha, amax);
  k_exp_sum<<<blkEH, 256, 0, stream>>>(dst_idx, amax, alpha, denom);
  k_att_scatter<<<EE / 4, 256, 0, stream>>>(src_idx, dst_idx, alpha, denom, x,
                                            att_out, accum);
  k_mean_bias<<<blkNC, 256, 0, stream>>>(accum, bias, batch_ptr, gsum, cnt);
  k_group_mean<<<blkGC, 256, 0, stream>>>(gsum, cnt);
  k_center_var<<<blkNC, 256, 0, stream>>>(accum, gsum, gn_scale, batch_ptr, gvar);
  k_group_inv<<<blkGC, 256, 0, stream>>>(gvar, cnt);
  k_norm_out<<<blkNC, 256, 0, stream>>>(accum, gvar, gn_w, gn_b, batch_ptr, y_out);
}